// NeRFRenderer_15006615733890
// MI455X (gfx1250) — compile-verified
//
#include <hip/hip_runtime.h>
#include <cstdint>

typedef __attribute__((ext_vector_type(2))) float v2f;
typedef __attribute__((ext_vector_type(8))) float v8f;
typedef __attribute__((ext_vector_type(4))) int   v4i;

#define NUM_S          64
#define CHUNK          8
#define NCHUNK         (NUM_S / CHUNK)              // 8
#define RAYS_PER_BLOCK 128
#define THREADS        256                          // 8 waves (wave32)
#define SIG_STRIDE     12                           // 8 floats + 4 pad: 16B-aligned rows, conflict-free
#define RGB_STRIDE     28                           // 24 floats + 4 pad
#define SIG_FLOATS     (RAYS_PER_BLOCK * SIG_STRIDE)   // 1536
#define RGB_FLOATS     (RAYS_PER_BLOCK * RGB_STRIDE)   // 3584
#define BUF_FLOATS     (SIG_FLOATS + RGB_FLOATS)       // 5120 floats -> 2 bufs = 40 KB LDS

#define T_STEP (4.0f / 63.0f)                       // linspace(2,6,64) step

// ---- CDNA5 async global->LDS copy (ASYNCcnt path) ----
__device__ __forceinline__ void async_copy_b128(void* lds, const void* gmem) {
#if __has_builtin(__builtin_amdgcn_global_load_async_to_lds_b128)
  __builtin_amdgcn_global_load_async_to_lds_b128(
      (__attribute__((address_space(1))) v4i*)gmem,
      (__attribute__((address_space(3))) v4i*)lds,
      0, 0);
#else
  asm volatile("global_load_async_to_lds_b128 %0, %1, off"
               :
               : "v"((uint32_t)(uint64_t)(__attribute__((address_space(3))) void*)lds),
                 "v"((uint64_t)gmem)
               : "memory");
#endif
}

#if __has_builtin(__builtin_amdgcn_s_wait_asynccnt)
#define WAIT_ASYNCCNT(n) __builtin_amdgcn_s_wait_asynccnt(n)
#else
#define WAIT_ASYNCCNT(n) asm volatile("s_wait_asynccnt %0" ::"n"(n))
#endif

__global__ __launch_bounds__(THREADS) void nerf_render_kernel(
    const float* __restrict__ rgb, const float* __restrict__ sigma,
    float* __restrict__ out_rgb, float* __restrict__ out_depth) {
  __shared__ __align__(16) float smem[2][BUF_FLOATS];

  const int tid  = threadIdx.x;
  const int wave = tid >> 5;
  const int lane = tid & 31;
  const int m    = lane & 15;   // ray-in-wave == WMMA M row
  const int half = lane >> 4;   // 0: samples k0,k0+1 ; 1: samples k0+2,k0+3

  const int R0   = blockIdx.x * RAYS_PER_BLOCK;
  const int rloc = wave * 16 + m;
  const int ray  = R0 + rloc;

  // per-chunk prefetch: each wave issues exactly 4 async b128 ops
  auto prefetch = [&](int c, float* buf) {
    {  // sigma: 128 rays x 8 floats = 256 b128 (one per thread)
      int r = tid >> 1, q = tid & 1;
      const float* g = sigma + ((size_t)(R0 + r) * NUM_S + c * CHUNK + q * 4);
      async_copy_b128(buf + r * SIG_STRIDE + q * 4, g);
    }
    float* rbuf = buf + SIG_FLOATS;
#pragma unroll
    for (int j = 0; j < 3; ++j) {  // rgb: 128 rays x 24 floats = 768 b128 (3/thread)
      int i = tid + j * THREADS;
      int r = i / 6, q = i % 6;
      const float* g = rgb + ((size_t)(R0 + r) * NUM_S + c * CHUNK) * 3 + q * 4;
      async_copy_b128(rbuf + r * RGB_STRIDE + q * 4, g);
    }
  };

  prefetch(0, smem[0]);

  float T  = 1.0f;                        // transmittance before current 4-sample group
  float aR = 0.f, aG = 0.f, aB = 0.f;
  v8f  dacc = {0.f, 0.f, 0.f, 0.f, 0.f, 0.f, 0.f, 0.f};

  // t(sample) handled incrementally: no divides/converts in the hot loop
  const float hoff = (float)half * (2.0f * T_STEP);  // this lane's K-half offset
  float tc = 2.0f;                                   // t at first sample of chunk

  for (int c = 0; c < NCHUNK; ++c) {
    if (c + 1 < NCHUNK) {
      prefetch(c + 1, smem[(c + 1) & 1]);
      WAIT_ASYNCCNT(4);            // only next chunk's 4 ops may remain in flight
    } else {
      WAIT_ASYNCCNT(0);
    }
    __syncthreads();

    const float* sb   = smem[c & 1];
    const float* srow = sb + rloc * SIG_STRIDE;
    const float* rrow = sb + SIG_FLOATS + rloc * RGB_STRIDE;

#pragma unroll
    for (int g = 0; g < CHUNK / 4; ++g) {
      const int kk = g * 4 + 2 * half;          // this lane's first sample of the group
      float s0 = srow[kk], s1 = srow[kk + 1];
      float e0 = __expf(-s0), e1 = __expf(-s1);
      float al0 = 1.0f - e0, al1 = 1.0f - e1;
      float f0 = e0 + 1e-10f, f1 = e1 + 1e-10f; // (1 - alpha + eps)
      float p  = f0 * f1;                       // local 2-sample transmittance factor
      float pOther = __shfl_xor(p, 16, 32);     // swap across K-halves (one ds op)
      // lower half: needs no prefix fixup; upper half: multiply by lower half's factor
      float pSel = half ? pOther : 1.0f;
      float Tloc = T * pSel;
      float w0 = al0 * Tloc;
      float w1 = al1 * (Tloc * f0);
      T *= p * pOther;                          // == pLow*pHigh for both halves

      const float* rp = rrow + kk * 3;
      aR += w0 * rp[0] + w1 * rp[3];
      aG += w0 * rp[1] + w1 * rp[4];
      aB += w0 * rp[2] + w1 * rp[5];

      // depth einsum via WMMA: A = W[16 rays x 4 samples], B[k][*] = t(k)
      float t0 = tc + ((float)(g * 4) * T_STEP + 0.0f) + hoff;  // consts fold after unroll
      v2f av, bv;
      av.x = w0;
      av.y = w1;
      bv.x = t0;
      bv.y = t0 + T_STEP;
      dacc = __builtin_amdgcn_wmma_f32_16x16x4_f32(false, av, false, bv,
                                                   (short)0, dacc, false, false);
    }
    __syncthreads();
    tc += (float)CHUNK * T_STEP;               // advance chunk t-base (one add)
  }

  // combine the two K-halves' rgb partials; half-0 lane of each ray writes
  float oR = __shfl_xor(aR, 16, 32);
  float oG = __shfl_xor(aG, 16, 32);
  float oB = __shfl_xor(aB, 16, 32);
  if (half == 0) {
    size_t o = (size_t)ray * 3;
    out_rgb[o + 0] = aR + oR;
    out_rgb[o + 1] = aG + oG;
    out_rgb[o + 2] = aB + oB;
  }

  // depth: D columns identical; VGPR v holds rows v (lanes 0-15) and v+8 (lanes 16-31)
  const int ray0 = R0 + wave * 16;
  if (lane == 0) {
#pragma unroll
    for (int v = 0; v < 8; ++v) out_depth[ray0 + v] = dacc[v];
  } else if (lane == 16) {
#pragma unroll
    for (int v = 0; v < 8; ++v) out_depth[ray0 + 8 + v] = dacc[v];
  }
}

extern "C" void kernel_launch(void* const* d_in, const int* in_sizes, int n_in,
                              void* d_out, int out_size, void* d_ws, size_t ws_size,
                              hipStream_t stream) {
  (void)n_in; (void)out_size; (void)d_ws; (void)ws_size;
  const float* rgb   = (const float*)d_in[0];   // [B, 64, 3] f32
  const float* sigma = (const float*)d_in[1];   // [B, 64]    f32
  const int Btot = in_sizes[1] / NUM_S;         // 262144
  float* out_rgb   = (float*)d_out;             // [B,3] then [B] depth
  float* out_depth = out_rgb + (size_t)Btot * 3;
  const int blocks = Btot / RAYS_PER_BLOCK;
  nerf_render_kernel<<<blocks, THREADS, 0, stream>>>(rgb, sigma, out_rgb, out_depth);
}